// VolumeRenderer_36172214567263
// MI455X (gfx1250) — compile-verified
//
#include <hip/hip_runtime.h>
#include <cstdint>

#define NUM_S 128
#define WAVES_PER_BLOCK 8

// CDNA5 async global->LDS path (ASYNCcnt). Guarded so either toolchain compiles.
#if defined(__HIP_DEVICE_COMPILE__) && defined(__gfx1250__) &&                 \
    __has_builtin(__builtin_amdgcn_global_load_async_to_lds_b128) &&           \
    __has_builtin(__builtin_amdgcn_s_wait_asynccnt)
#define USE_ASYNC_LDS 1
#else
#define USE_ASYNC_LDS 0
#endif

__device__ __forceinline__ float fast_rcp(float x) {
#if defined(__HIP_DEVICE_COMPILE__) && __has_builtin(__builtin_amdgcn_rcpf)
    return __builtin_amdgcn_rcpf(x);
#else
    return 1.0f / x;
#endif
}

__device__ __forceinline__ float fast_sigmoid(float x) {
    return fast_rcp(1.0f + __expf(-x));  // v_exp_f32 + v_rcp_f32
}

__global__ __launch_bounds__(256) void volrender_kernel(
    const float* __restrict__ rf,      // [R,128,4]
    const float* __restrict__ depth,   // [R,128]
    const float* __restrict__ dirs,    // [R,3]
    float* __restrict__ rgb_map,       // [R,3]
    float* __restrict__ disp_map,      // [R]
    float* __restrict__ acc_map,       // [R]
    float* __restrict__ weights,       // [R,128]
    float* __restrict__ depth_map,     // [R]
    int R)
{
    const int lane = threadIdx.x & 31;
    const int wv   = threadIdx.x >> 5;
    const int ray  = blockIdx.x * WAVES_PER_BLOCK + wv;

#if USE_ASYNC_LDS
    __shared__ float4 lbuf[WAVES_PER_BLOCK][4][32];   // 16 KB / block
#endif

    if (ray >= R) return;   // uniform per wave: shuffles below see full EXEC

    const float* rf_ray = rf + (size_t)ray * (NUM_S * 4);

#if USE_ASYNC_LDS
    // Builtin signature (from compiler diagnostic): first param is
    // 'int __attribute__((vector_size(16))) addrspace(1)*' -> match it exactly.
    typedef int v4i_t __attribute__((vector_size(16)));
    typedef __attribute__((address_space(1))) v4i_t* gv4i_p;
    typedef __attribute__((address_space(3))) v4i_t* lv4i_p;
    {
        const float* g = rf_ray + lane * 4;   // float4 per lane, 512B per chunk per wave
        #define ISSUE_ASYNC(c)                                                   \
            __builtin_amdgcn_global_load_async_to_lds_b128(                      \
                (gv4i_p)(g + (c) * 128),                                         \
                (lv4i_p)(&lbuf[wv][(c)][lane]), 0, 0)
        ISSUE_ASYNC(0); ISSUE_ASYNC(1); ISSUE_ASYNC(2); ISSUE_ASYNC(3);
        #undef ISSUE_ASYNC
    }
#endif

    // ---- ray-direction norm (tiny, uniform per wave) ----
    const float dx = dirs[ray * 3 + 0];
    const float dy = dirs[ray * 3 + 1];
    const float dz = dirs[ray * 3 + 2];
    const float dnorm = sqrtf(dx * dx + dy * dy + dz * dz);

    // ---- depth values: 4 coalesced chunks; dists via lane shuffles (single read of depth) ----
    float dep[4], dst[4];
    #pragma unroll
    for (int c = 0; c < 4; ++c)
        dep[c] = depth[ray * NUM_S + c * 32 + lane];

    #pragma unroll
    for (int c = 0; c < 4; ++c) {
        float nxt = __shfl_down(dep[c], 1, 32);
        if (c < 3) {
            float head = __shfl(dep[c + 1], 0, 32);
            nxt = (lane == 31) ? head : nxt;
            dst[c] = (nxt - dep[c]) * dnorm;
        } else {
            dst[c] = (lane == 31) ? (1e10f * dnorm) : ((nxt - dep[c]) * dnorm);
        }
    }

#if USE_ASYNC_LDS
    __builtin_amdgcn_s_wait_asynccnt(0);   // all 4 chunks resident in LDS
#endif

    float T   = 1.0f;                       // running transmittance carry
    float s_r = 0.f, s_g = 0.f, s_b = 0.f, s_d = 0.f, s_a = 0.f;

    #pragma unroll
    for (int c = 0; c < 4; ++c) {
#if USE_ASYNC_LDS
        float4 v = lbuf[wv][c][lane];                              // ds_load_b128
#else
        float4 v = *(const float4*)(rf_ray + c * 128 + lane * 4);  // global_load_b128
#endif
        float r = fast_sigmoid(v.x);
        float g = fast_sigmoid(v.y);
        float b = fast_sigmoid(v.z);
        float sigma = fmaxf(v.w, 0.0f);
        float alpha = 1.0f - __expf(-sigma * dst[c]);
        float t     = 1.0f - alpha + 1e-10f;

        // inclusive multiplicative Hillis-Steele scan within the 32-sample chunk
        float incl = t;
        #pragma unroll
        for (int d = 1; d < 32; d <<= 1) {
            float o = __shfl_up(incl, d, 32);
            incl *= (lane >= d) ? o : 1.0f;
        }
        float excl = __shfl_up(incl, 1, 32);
        excl = (lane == 0) ? 1.0f : excl;

        float w = alpha * (T * excl);        // exclusive cumprod * alpha
        T *= __shfl(incl, 31, 32);           // carry whole-chunk product

        weights[ray * NUM_S + c * 32 + lane] = w;   // coalesced b32 store
        s_r += w * r;
        s_g += w * g;
        s_b += w * b;
        s_d += w * dep[c];
        s_a += w;
    }

    // ---- wave32 butterfly reductions for the per-ray maps ----
    #pragma unroll
    for (int d = 16; d >= 1; d >>= 1) {
        s_r += __shfl_xor(s_r, d, 32);
        s_g += __shfl_xor(s_g, d, 32);
        s_b += __shfl_xor(s_b, d, 32);
        s_d += __shfl_xor(s_d, d, 32);
        s_a += __shfl_xor(s_a, d, 32);
    }

    if (lane == 0) {
        rgb_map[ray * 3 + 0] = s_r;
        rgb_map[ray * 3 + 1] = s_g;
        rgb_map[ray * 3 + 2] = s_b;
        depth_map[ray] = s_d;
        acc_map[ray]   = s_a;
        disp_map[ray]  = 1.0f / fmaxf(1e-10f, s_d / s_a);
    }
}

extern "C" void kernel_launch(void* const* d_in, const int* in_sizes, int n_in,
                              void* d_out, int out_size, void* d_ws, size_t ws_size,
                              hipStream_t stream) {
    (void)n_in; (void)out_size; (void)d_ws; (void)ws_size;

    const float* rf    = (const float*)d_in[0];   // radiance_field [R,128,4]
    const float* depth = (const float*)d_in[1];   // depth_values   [R,128]
    const float* dirs  = (const float*)d_in[2];   // ray_directions [R,3]

    const int R = in_sizes[1] / NUM_S;

    float* out       = (float*)d_out;
    float* rgb_map   = out;                               // [R,3]
    float* disp_map  = rgb_map + (size_t)R * 3;           // [R]
    float* acc_map   = disp_map + R;                      // [R]
    float* wts       = acc_map + R;                       // [R,128]
    float* depth_map = wts + (size_t)R * NUM_S;           // [R]

    const int blocks = (R + WAVES_PER_BLOCK - 1) / WAVES_PER_BLOCK;
    volrender_kernel<<<blocks, 32 * WAVES_PER_BLOCK, 0, stream>>>(
        rf, depth, dirs, rgb_map, disp_map, acc_map, wts, depth_map, R);
}